// SegmentorNew_35296041238601
// MI455X (gfx1250) — compile-verified
//
#include <hip/hip_runtime.h>

#define V_  12
#define N_  80000
#define C_  256
#define H_  128
#define W_  128
#define G_  4096
#define M_  512
#define E_  200000
#define L_  20
#define C2_ 128  // C/2

typedef __attribute__((ext_vector_type(2))) float v2f;
typedef __attribute__((ext_vector_type(8))) float v8f;

__global__ void k_zero(float* __restrict__ p, int n) {
  int i = blockIdx.x * blockDim.x + threadIdx.x;
  if (i < n) p[i] = 0.0f;
}

// group_feat[g][c] = sum_v w(v,n) * img[v][c][y][x] / (sum_v w + 1e-6), n = centers[g]
__global__ void k_group_gather(const float* __restrict__ img,
                               const float* __restrict__ is_seen,
                               const int*   __restrict__ coords,
                               const int*   __restrict__ centers,
                               float* __restrict__ group_feat) {
  int g = blockIdx.x;    // 4096
  int c = threadIdx.x;   // 256
  int n = centers[g];
  float acc = 0.0f, den = 0.0f;
  for (int v = 0; v < V_; ++v) {
    int x = coords[((size_t)v * N_ + n) * 2 + 0];
    int y = coords[((size_t)v * N_ + n) * 2 + 1];
    float w = is_seen[(size_t)v * N_ + n];
    den += w;
    acc += w * img[(((size_t)v * C_ + c) * H_ + y) * W_ + x];
  }
  group_feat[(size_t)g * C_ + c] = acc / (den + 1e-6f);
}

// scatter group_feat rows into per-mask sums/counts
__global__ void k_mask_scatter(const float* __restrict__ group_feat,
                               const int*   __restrict__ group2mask,
                               float* __restrict__ msum, float* __restrict__ mcnt) {
  int g = blockIdx.x;    // 4096
  int c = threadIdx.x;   // 256
  int m = group2mask[g];
  unsafeAtomicAdd(&msum[(size_t)m * C_ + c], group_feat[(size_t)g * C_ + c]);
  if (c == 0) unsafeAtomicAdd(&mcnt[m], 1.0f);
}

__global__ void k_mask_norm(const float* __restrict__ msum,
                            const float* __restrict__ mcnt,
                            float* __restrict__ mask_feat) {
  int i = blockIdx.x * blockDim.x + threadIdx.x;  // 512*256
  int m = i >> 8;
  float cc = mcnt[m];
  mask_feat[i] = (cc > 0.0f) ? msum[i] / fmaxf(cc, 1.0f) : 0.0f;
}

// mask_h1[512][128] = mask_feat[512][256] @ W1[256][128]   (fp32 WMMA)
__global__ void k_mask_h1(const float* __restrict__ mask_feat,
                          const float* __restrict__ W1,
                          float* __restrict__ mask_h1) {
  int lane = threadIdx.x & 31;
  int wid  = threadIdx.x >> 5;
  int tile = blockIdx.x * 4 + wid;          // 0..255 (32 M-tiles x 8 N-tiles)
  int mt = tile >> 3, nt = tile & 7;
  int mrow = (mt << 4) + (lane & 15);
  int ncol = (nt << 4) + (lane & 15);
  int koff = (lane >> 4) << 1;              // 0 or 2
  v8f c = {};
  for (int kk = 0; kk < C_; kk += 4) {
    v2f a, b;
    int k0 = kk + koff, k1 = kk + koff + 1;
    a.x = mask_feat[(size_t)mrow * C_ + k0];
    a.y = mask_feat[(size_t)mrow * C_ + k1];
    b.x = W1[(size_t)k0 * C2_ + ncol];
    b.y = W1[(size_t)k1 * C2_ + ncol];
    c = __builtin_amdgcn_wmma_f32_16x16x4_f32(false, a, false, b, (short)0, c, false, false);
  }
  for (int i = 0; i < 8; ++i) {
    int r = (mt << 4) + i + ((lane >> 4) << 3);
    mask_h1[(size_t)r * C2_ + ncol] = c[i];
  }
}

// fsum[p][k] += mask_h1[pair_mask[e]][k]; cnt[p] += 1
__global__ void k_edge_scatter(const float* __restrict__ mask_h1,
                               const int*   __restrict__ pair_mask,
                               const int*   __restrict__ pair_point,
                               float* __restrict__ fsum, float* __restrict__ cnt) {
  long long idx = (long long)blockIdx.x * blockDim.x + threadIdx.x;
  if (idx >= (long long)E_ * C2_) return;
  int e = (int)(idx >> 7);
  int k = (int)(idx & 127);
  int m = pair_mask[e];
  int p = pair_point[e];
  unsafeAtomicAdd(&fsum[(size_t)p * C2_ + k], mask_h1[(size_t)m * C2_ + k]);
  if (k == 0) unsafeAtomicAdd(&cnt[p], 1.0f);
}

// logits[N][20] = relu(fsum/max(cnt,1e-6) + b1) @ W2[128][20] + b2
// relu/bias/scale fused into the WMMA A-fragment load; N padded 20->32 (2 col tiles).
__global__ void k_final(const float* __restrict__ fsum, const float* __restrict__ cnt,
                        const float* __restrict__ b1,   const float* __restrict__ W2,
                        const float* __restrict__ b2,   float* __restrict__ out) {
  int lane = threadIdx.x & 31;
  int wid  = threadIdx.x >> 5;
  int m_base = ((blockIdx.x << 3) + wid) << 4;     // 16 points per wave
  int m = m_base + (lane & 15);                    // this lane's A-matrix row
  bool valid = (m < N_);
  float scale = 0.0f;
  if (valid) scale = 1.0f / fmaxf(cnt[m], 1e-6f);
  int koff = (lane >> 4) << 1;
  int col  = lane & 15;
  v8f c0 = {}, c1 = {};
  for (int kk = 0; kk < C2_; kk += 4) {
    int k0 = kk + koff, k1 = kk + koff + 1;
    v2f a, b0, b1v;
    if (valid) {
      a.x = fmaxf(fsum[(size_t)m * C2_ + k0] * scale + b1[k0], 0.0f);
      a.y = fmaxf(fsum[(size_t)m * C2_ + k1] * scale + b1[k1], 0.0f);
    } else {
      a.x = 0.0f; a.y = 0.0f;
    }
    b0.x = W2[(size_t)k0 * L_ + col];               // col < 16 < 20 always valid
    b0.y = W2[(size_t)k1 * L_ + col];
    int col1 = 16 + col;
    b1v.x = (col1 < L_) ? W2[(size_t)k0 * L_ + col1] : 0.0f;
    b1v.y = (col1 < L_) ? W2[(size_t)k1 * L_ + col1] : 0.0f;
    c0 = __builtin_amdgcn_wmma_f32_16x16x4_f32(false, a, false, b0,  (short)0, c0, false, false);
    c1 = __builtin_amdgcn_wmma_f32_16x16x4_f32(false, a, false, b1v, (short)0, c1, false, false);
  }
  for (int i = 0; i < 8; ++i) {
    int r = m_base + i + ((lane >> 4) << 3);
    if (r < N_) {
      out[(size_t)r * L_ + col] = c0[i] + b2[col];
      int col1 = 16 + col;
      if (col1 < L_) out[(size_t)r * L_ + col1] = c1[i] + b2[col1];
    }
  }
}

extern "C" void kernel_launch(void* const* d_in, const int* in_sizes, int n_in,
                              void* d_out, int out_size, void* d_ws, size_t ws_size,
                              hipStream_t stream) {
  const float* img        = (const float*)d_in[0];
  const float* is_seen    = (const float*)d_in[1];
  const float* W1         = (const float*)d_in[2];
  const float* b1         = (const float*)d_in[3];
  const float* W2         = (const float*)d_in[4];
  const float* b2         = (const float*)d_in[5];
  const int*   coords     = (const int*)d_in[6];
  const int*   centers    = (const int*)d_in[7];
  const int*   group2mask = (const int*)d_in[8];
  const int*   pair_mask  = (const int*)d_in[9];
  const int*   pair_point = (const int*)d_in[10];
  float* out = (float*)d_out;

  // workspace layout (floats)
  float* ws         = (float*)d_ws;
  float* group_feat = ws;                               // 4096*256
  float* msum       = group_feat + (size_t)G_ * C_;     // 512*256
  float* mcnt       = msum + (size_t)M_ * C_;           // 512
  float* mask_feat  = mcnt + M_;                        // 512*256
  float* mask_h1    = mask_feat + (size_t)M_ * C_;      // 512*128
  float* fsum       = mask_h1 + (size_t)M_ * C2_;       // 80000*128
  // cnt = fsum + N_*C2_ (contiguous)                   // 80000

  // zero only the atomic-accumulated regions (others are fully overwritten)
  int nz1 = M_ * C_ + M_;                               // msum + mcnt
  k_zero<<<(nz1 + 255) / 256, 256, 0, stream>>>(msum, nz1);
  int nz2 = N_ * C2_ + N_;                              // fsum + cnt
  k_zero<<<(nz2 + 255) / 256, 256, 0, stream>>>(fsum, nz2);

  k_group_gather<<<G_, C_, 0, stream>>>(img, is_seen, coords, centers, group_feat);
  k_mask_scatter<<<G_, C_, 0, stream>>>(group_feat, group2mask, msum, mcnt);
  k_mask_norm<<<(M_ * C_) / 256, 256, 0, stream>>>(msum, mcnt, mask_feat);
  k_mask_h1<<<64, 128, 0, stream>>>(mask_feat, W1, mask_h1);

  long long ne = (long long)E_ * C2_;
  k_edge_scatter<<<(int)((ne + 255) / 256), 256, 0, stream>>>(
      mask_h1, pair_mask, pair_point, fsum, fsum + (size_t)N_ * C2_);

  int tiles = (N_ + 15) / 16;                           // 5000
  k_final<<<(tiles + 7) / 8, 256, 0, stream>>>(
      fsum, fsum + (size_t)N_ * C2_, b1, W2, b2, out);
}